// VectorQuantizerEMA_83373905150272
// MI455X (gfx1250) — compile-verified
//
#include <hip/hip_runtime.h>
#include <hip/hip_bf16.h>

typedef __attribute__((ext_vector_type(2))) float v2f;
typedef __attribute__((ext_vector_type(4))) float v4f;
typedef __attribute__((ext_vector_type(8))) float v8f;
typedef __attribute__((ext_vector_type(4))) unsigned int u32x4;
typedef __attribute__((ext_vector_type(8))) int i32x8;
typedef __attribute__((ext_vector_type(4))) int i32x4;

#define K_CODES 4096
#define D_DIM   256
#define B_ROWS  32768
#define BETA_C  0.25f
#define DECAY_C 0.99f
#define EPS_C   1e-5f

#define MT  128       // rows (z_e) per block
#define NT  64        // codes per LDS tile
#define LDZ 260       // padded LDS row stride (floats): 260%64==4 -> conflict-free
#define LDC 260
#define NTILES (K_CODES / NT)

#if defined(__has_builtin)
#if __has_builtin(__builtin_amdgcn_tensor_load_to_lds) && __has_builtin(__builtin_amdgcn_s_wait_tensorcnt)
#define HAVE_TDM 1
#endif
#endif
#ifndef HAVE_TDM
#define HAVE_TDM 0
#endif

// ---------------------------------------------------------------- zero ws
__global__ void vq_zero_kernel(float* __restrict__ p, int n) {
    int i = blockIdx.x * blockDim.x + threadIdx.x;
    if (i < n) p[i] = 0.0f;
}

// ---------------------------------------------------------------- ||c||^2
__global__ void vq_cnorm_kernel(const float* __restrict__ cb, float* __restrict__ cn) {
    int k = blockIdx.x * blockDim.x + threadIdx.x;
    if (k < K_CODES) {
        const float* r = cb + (size_t)k * D_DIM;
        float s = 0.0f;
        for (int d = 0; d < D_DIM; ++d) { float v = r[d]; s += v * v; }
        cn[k] = s;
    }
}

#if HAVE_TDM
// One TDM descriptor: DMA a NT x 256 f32 tile (row-major) into LDS with a
// 4-DWORD pad after every 256-DWORD row -> LDS row stride 260 floats.
__device__ __forceinline__ void tdm_load_tile(const float* lds_dst, const float* gsrc) {
    unsigned long long ga = (unsigned long long)(size_t)gsrc;
    u32x4 g0;
    g0[0] = 1u;                                             // count=1 (valid user D#)
    g0[1] = (unsigned)(size_t)lds_dst;                      // LDS byte offset
    g0[2] = (unsigned)ga;                                   // global_addr[31:0]
    g0[3] = (unsigned)((ga >> 32) & 0x1FFFFFFu) | (2u << 30); // addr[56:32] | type=2
    i32x8 g1;
    g1[0] = (2 << 16)          // data_size = 4 bytes
          | (1 << 20)          // pad_enable
          | (7 << 22)          // pad_interval = 256 DWORDs (one row)
          | (3 << 25);         // pad_amount   = 4 DWORDs  (stride 260)
    g1[1] = (int)(D_DIM << 16);    // tensor_dim0 = 256      (bits 63:48)
    g1[2] = (int)(K_CODES << 16);  // tensor_dim1 = 4096     (bits 95:80)
    g1[3] = (int)(D_DIM << 16);    // tile_dim0   = 256      (bits 127:112)
    g1[4] = NT;                    // tile_dim1   = 64       (bits 143:128)
    g1[5] = D_DIM;                 // tensor_dim0_stride=256 (bits 191:160)
    g1[6] = 0;
    g1[7] = 0;
    i32x4 gz4 = {0, 0, 0, 0};
    i32x8 gz8 = {0, 0, 0, 0, 0, 0, 0, 0};
    // clang-23 / therock form: (g0, g1, g2, g3, g4, cpol)
    __builtin_amdgcn_tensor_load_to_lds(g0, g1, gz4, gz4, gz8, 0);
}
#endif

// ------------------------------------------------ fused GEMM + argmin (WMMA f32)
__global__ __launch_bounds__(256)
void vq_dist_argmin_kernel(const float* __restrict__ z_e,
                           const float* __restrict__ cb,
                           const float* __restrict__ c_norm,
                           float* __restrict__ idx_f,     // d_out idx region
                           int*   __restrict__ idx_i,     // ws int idx
                           float* __restrict__ counts,    // ws [K]
                           float* __restrict__ commit_sum)// ws scalar
{
    extern __shared__ float smem[];
    float* zs   = smem;                    // MT*LDZ
    float* csh0 = zs   + MT * LDZ;         // NT*LDC  (ping)
    float* csh1 = csh0 + NT * LDC;         // NT*LDC  (pong)
    float* cn   = csh1 + NT * LDC;         // NT
    float* en   = cn   + NT;               // MT
    float* bsum = en   + MT;               // 1

    const int tid  = threadIdx.x;
    const int lane = tid & 31;
    const int wave = tid >> 5;             // 8 waves: one 16-row strip each
    const int half = lane >> 4;            // WMMA lane-half
    const int l16  = lane & 15;
    const int row0 = blockIdx.x * MT;

#if HAVE_TDM
    // Kick the first codebook tile DMA before anything else; it overlaps with
    // the z-panel staging below.  TENSORcnt tracks completion.
    if (wave == 0) tdm_load_tile(csh0, cb);
#endif

    // Stage the whole 128x256 f32 z panel once (130KB of CDNA5's 320KB LDS),
    // 16B-vectorized: LDZ=260 keeps every row 16B-aligned.
    for (int i = tid; i < MT * (D_DIM / 4); i += 256) {
        int r = i >> 6;             // 64 float4 per row
        int c = (i & 63) << 2;
        *(v4f*)&zs[r * LDZ + c] = *(const v4f*)&z_e[(size_t)(row0 + r) * D_DIM + c];
    }
    if (tid == 0) bsum[0] = 0.0f;
    __syncthreads();

    // e_norm per row (needed only for the commit-loss value)
    if (tid < MT) {
        const float* zr = zs + tid * LDZ;
        float s = 0.0f;
        for (int d = 0; d < D_DIM; ++d) { float v = zr[d]; s += v * v; }
        en[tid] = s;
    }

    float minv[8];
    int   mini[8];
#pragma unroll
    for (int r = 0; r < 8; ++r) { minv[r] = 3.4e38f; mini[r] = 0; }

    // A operand: lane (l16) -> row, lane-half -> K pair {0,1}/{2,3}
    const int   arow = wave * 16 + l16;
    const float* zrow = zs + arow * LDZ + half * 2;

    for (int it = 0; it < NTILES; ++it) {
        const int n0 = it * NT;
        float* cur = (it & 1) ? csh1 : csh0;

#if HAVE_TDM
        if (wave == 0) {
            if (it + 1 < NTILES) {            // prefetch next tile into the other buffer
                tdm_load_tile((it & 1) ? csh0 : csh1, cb + (size_t)(n0 + NT) * D_DIM);
                __builtin_amdgcn_s_wait_tensorcnt(1);  // wait for *current* tile only
            } else {
                __builtin_amdgcn_s_wait_tensorcnt(0);
            }
        }
#else
        for (int i = tid; i < NT * (D_DIM / 4); i += 256) {
            int r = i >> 6;
            int c = (i & 63) << 2;
            *(v4f*)&cur[r * LDC + c] = *(const v4f*)&cb[(size_t)(n0 + r) * D_DIM + c];
        }
#endif
        if (tid < NT) cn[tid] = c_norm[n0 + tid];
        __syncthreads();

        v8f acc[4];
#pragma unroll
        for (int t = 0; t < 4; ++t) acc[t] = (v8f){0.f,0.f,0.f,0.f,0.f,0.f,0.f,0.f};

        const float* crow = cur + l16 * LDC + half * 2;

        // Software-pipelined K-loop: prefetch k-step d0+4 while the 4 WMMAs of
        // d0 issue back-to-back (no per-WMMA s_wait_dscnt 0).
        v2f a_c, b_c[4];
        a_c.x = zrow[0]; a_c.y = zrow[1];
#pragma unroll
        for (int t = 0; t < 4; ++t) {
            const float* p = crow + t * 16 * LDC;
            b_c[t].x = p[0]; b_c[t].y = p[1];
        }
#pragma unroll 4
        for (int d0 = 0; d0 < D_DIM - 4; d0 += 4) {
            v2f a_n, b_n[4];
            a_n.x = zrow[d0 + 4]; a_n.y = zrow[d0 + 5];
#pragma unroll
            for (int t = 0; t < 4; ++t) {
                const float* p = crow + t * 16 * LDC + d0 + 4;
                b_n[t].x = p[0]; b_n[t].y = p[1];
            }
#pragma unroll
            for (int t = 0; t < 4; ++t)
                acc[t] = __builtin_amdgcn_wmma_f32_16x16x4_f32(
                    false, a_c, false, b_c[t], (short)0, acc[t], false, false);
            a_c = a_n;
#pragma unroll
            for (int t = 0; t < 4; ++t) b_c[t] = b_n[t];
        }
#pragma unroll
        for (int t = 0; t < 4; ++t)
            acc[t] = __builtin_amdgcn_wmma_f32_16x16x4_f32(
                false, a_c, false, b_c[t], (short)0, acc[t], false, false);

        // Candidate distances: ||c||^2 - 2 z.c  (e_norm constant per row -> no
        // effect on argmin; added back only for the commit loss at the end).
#pragma unroll
        for (int t = 0; t < 4; ++t) {
            float cnv  = cn[t * 16 + l16];
            int   gcol = n0 + t * 16 + l16;
#pragma unroll
            for (int r = 0; r < 8; ++r) {
                float dist = cnv - 2.0f * acc[t][r];
                if (dist < minv[r]) { minv[r] = dist; mini[r] = gcol; } // strict < keeps first idx
            }
        }
        __syncthreads();   // all waves done with `cur` before it is re-DMA'd
    }

    // Cross-lane min+argmin inside each 16-lane half (rows r and r+8).
    float cpart = 0.0f;
#pragma unroll
    for (int r = 0; r < 8; ++r) {
        float v  = minv[r];
        int   ix = mini[r];
#pragma unroll
        for (int off = 8; off >= 1; off >>= 1) {
            float v2 = __shfl_xor(v, off, 32);
            int   i2 = __shfl_xor(ix, off, 32);
            if (v2 < v || (v2 == v && i2 < ix)) { v = v2; ix = i2; }
        }
        if (l16 == 0) {
            int rloc = wave * 16 + half * 8 + r;
            int grow = row0 + rloc;
            idx_f[grow] = (float)ix;
            idx_i[grow] = ix;
            atomicAdd(&counts[ix], 1.0f);
            cpart += en[rloc] + v;                // ||z-c||^2 at the argmin
        }
    }
    atomicAdd(&bsum[0], cpart);                   // LDS reduce
    __syncthreads();
    if (tid == 0) atomicAdd(commit_sum, bsum[0]); // one global atomic per block
}

// ------------------------------------------- z_q gather + dw index_add scatter
__global__ void vq_gather_scatter_kernel(const float* __restrict__ z_e,
                                         const float* __restrict__ cb,
                                         const int*   __restrict__ idx_i,
                                         float* __restrict__ zq_out,
                                         float* __restrict__ dw) {
    size_t g = (size_t)blockIdx.x * 256 + threadIdx.x;  // B*D threads
    int b = (int)(g >> 8);
    int d = (int)(g & 255);
    int k = idx_i[b];
    zq_out[g] = cb[(size_t)k * D_DIM + d];              // z_q_ste == z_q numerically
    atomicAdd(&dw[(size_t)k * D_DIM + d], z_e[g]);
}

// ------------------------------------------------- EMA cluster sizes + n
__global__ void vq_cluster_kernel(const float* __restrict__ ema_cluster,
                                  const float* __restrict__ counts,
                                  float* __restrict__ out_cluster,
                                  float* __restrict__ n_sum) {
    int k = blockIdx.x * blockDim.x + threadIdx.x;
    if (k < K_CODES) {
        float nc = ema_cluster[k] * DECAY_C + counts[k] * (1.0f - DECAY_C);
        out_cluster[k] = nc;
        atomicAdd(n_sum, nc);
    }
}

// ----------------------------------- EMA weights, normalized codebook, commit
__global__ void vq_codebook_kernel(const float* __restrict__ ema_weight,
                                   const float* __restrict__ dw,
                                   const float* __restrict__ out_cluster,
                                   const float* __restrict__ n_sum,
                                   const float* __restrict__ commit_sum,
                                   float* __restrict__ out_codebook,
                                   float* __restrict__ out_ema,
                                   float* __restrict__ out_commit) {
    size_t g = (size_t)blockIdx.x * 256 + threadIdx.x;  // K*D threads
    int k = (int)(g >> 8);
    float n  = *n_sum;
    float cs = (out_cluster[k] + EPS_C) / (n + (float)K_CODES * EPS_C) * n;
    float e  = ema_weight[g] * DECAY_C + dw[g] * (1.0f - DECAY_C);
    out_ema[g]      = e;
    out_codebook[g] = e / cs;
    if (g == 0)
        *out_commit = BETA_C * (*commit_sum) / ((float)B_ROWS * (float)D_DIM);
}

extern "C" void kernel_launch(void* const* d_in, const int* in_sizes, int n_in,
                              void* d_out, int out_size, void* d_ws, size_t ws_size,
                              hipStream_t stream) {
    (void)in_sizes; (void)n_in; (void)out_size; (void)ws_size;
    const float* z_e         = (const float*)d_in[0];   // [B,D]
    const float* codebook    = (const float*)d_in[1];   // [K,D]
    const float* ema_cluster = (const float*)d_in[2];   // [K]
    const float* ema_weight  = (const float*)d_in[3];   // [K,D]

    // d_out: z_q_ste[B*D] | idx[B] | commit[1] | codebook[K*D] | cluster[K] | ema[K*D]
    float* out        = (float*)d_out;
    float* o_zq       = out;
    float* o_idx      = o_zq + (size_t)B_ROWS * D_DIM;
    float* o_commit   = o_idx + B_ROWS;
    float* o_codebook = o_commit + 1;
    float* o_cluster  = o_codebook + (size_t)K_CODES * D_DIM;
    float* o_ema      = o_cluster + K_CODES;

    // ws: counts[K] | dw[K*D] | commit | n | c_norm[K] | idx_i[B]  (~4.4 MB)
    float* w        = (float*)d_ws;
    float* w_counts = w;
    float* w_dw     = w_counts + K_CODES;
    float* w_commit = w_dw + (size_t)K_CODES * D_DIM;
    float* w_nsum   = w_commit + 1;
    float* w_cnorm  = w_nsum + 1;
    int*   w_idx    = (int*)(w_cnorm + K_CODES);

    const int nzero = K_CODES + K_CODES * D_DIM + 2;    // counts+dw+commit+n contiguous
    vq_zero_kernel<<<(nzero + 255) / 256, 256, 0, stream>>>(w, nzero);

    vq_cnorm_kernel<<<K_CODES / 256, 256, 0, stream>>>(codebook, w_cnorm);

    const size_t smem_bytes =
        (size_t)(MT * LDZ + 2 * NT * LDC + NT + MT + 1) * sizeof(float); // ~261KB
    vq_dist_argmin_kernel<<<B_ROWS / MT, 256, smem_bytes, stream>>>(
        z_e, codebook, w_cnorm, o_idx, w_idx, w_counts, w_commit);

    vq_gather_scatter_kernel<<<(B_ROWS * D_DIM) / 256, 256, 0, stream>>>(
        z_e, codebook, w_idx, o_zq, w_dw);

    vq_cluster_kernel<<<K_CODES / 256, 256, 0, stream>>>(
        ema_cluster, w_counts, o_cluster, w_nsum);

    vq_codebook_kernel<<<(K_CODES * D_DIM) / 256, 256, 0, stream>>>(
        ema_weight, w_dw, o_cluster, w_nsum, w_commit, o_codebook, o_ema, o_commit);
}